// GeodesicWindowPartition_54262616818062
// MI455X (gfx1250) — compile-verified
//
#include <hip/hip_runtime.h>
#include <hip/hip_bf16.h>
#include <stdint.h>

// Problem constants (fixed by the reference)
#define NB 8        // batch
#define NN 40962    // points
#define NC 128      // channels
#define NW 162      // windows
#define CHUNK 256
#define NCHUNKS ((NN + CHUNK - 1) / CHUNK)   // 161

// ---- CDNA5 async global<->LDS path (gfx1250) -------------------------------
#if defined(__has_builtin)
#if __has_builtin(__builtin_amdgcn_global_load_async_to_lds_b128) && \
    __has_builtin(__builtin_amdgcn_global_store_async_from_lds_b128) && \
    __has_builtin(__builtin_amdgcn_s_wait_asynccnt)
#define USE_ASYNC_LDS 1
#endif
#endif

typedef int v4i __attribute__((ext_vector_type(4)));          // builtin arg0: v4i* (global side)
typedef __attribute__((address_space(3))) v4i lds_v4i_t;      // builtin arg1: AS3 v4i* (LDS side)

// ---- Kernel 5 (defined first so its asm shows in the disasm snippet):
// the big scatter copy via async global->LDS->global DMA ---------------------
__global__ void k_copy(const float* __restrict__ x, const int* __restrict__ dest_row,
                       float* __restrict__ out, int M) {
  __shared__ float lds[8 * NC];                    // 8 waves * 512B rows = 4 KB
  const int wave = threadIdx.x >> 5;
  const int lane = threadIdx.x & 31;
  const int row = blockIdx.x * 8 + wave;           // [0, NB*NN), exact
  const int b = row / NN;
  const int i = row - b * NN;
  const size_t src_off = ((size_t)b * NN + (size_t)i) * NC + (size_t)lane * 4;
  const int drow = dest_row[i];
  const size_t dst_off = ((size_t)b * (size_t)(NW) * (size_t)M + (size_t)drow) * NC
                         + (size_t)lane * 4;
#if USE_ASYNC_LDS
  lds_v4i_t* lp = (lds_v4i_t*)&lds[wave * NC + lane * 4];
  __builtin_amdgcn_global_load_async_to_lds_b128((v4i*)(x + src_off), lp, 0, 0);
  __builtin_amdgcn_s_wait_asynccnt(0);             // LDS row is now resident
  __builtin_amdgcn_global_store_async_from_lds_b128((v4i*)(out + dst_off), lp, 0, 0);
  // s_endpgm performs an implicit wait-idle, draining ASYNCcnt
#else
  const float4 v = *reinterpret_cast<const float4*>(x + src_off);
  *reinterpret_cast<float4*>(out + dst_off) = v;
  (void)lds;
#endif
}

// ---- Kernel 1: per-chunk histograms ----------------------------------------
__global__ void k_hist(const int* __restrict__ wid, int* __restrict__ partial) {
  __shared__ int lh[NW];
  const int t = threadIdx.x;
  if (t < NW) lh[t] = 0;
  __syncthreads();
  const int i = blockIdx.x * CHUNK + t;
  if (i < NN) atomicAdd(&lh[wid[i]], 1);
  __syncthreads();
  if (t < NW) partial[blockIdx.x * NW + t] = lh[t];
}

// ---- Kernel 2: scans (per-window over chunks; per-window exclusive offsets) -
__global__ void k_scan(const int* __restrict__ partial, int* __restrict__ counts,
                       int* __restrict__ offsets, int* __restrict__ chunk_base,
                       float* __restrict__ counts_out) {
  __shared__ int sc[NW];
  const int w = threadIdx.x;
  if (w < NW) {
    int run = 0;
    for (int c = 0; c < NCHUNKS; ++c) {
      chunk_base[c * NW + w] = run;
      run += partial[c * NW + w];
    }
    sc[w] = run;
    counts[w] = run;
  }
  __syncthreads();
  if (threadIdx.x == 0) {
    int acc = 0;
    for (int ww = 0; ww < NW; ++ww) { offsets[ww] = acc; acc += sc[ww]; }
  }
  __syncthreads();
  if (w < NW) {
    const int off = offsets[w];
    for (int c = 0; c < NCHUNKS; ++c) chunk_base[c * NW + w] += off;
    counts_out[w] = (float)sc[w];
  }
}

// ---- Kernel 3: stable global rank, order output, destination row ----------
__global__ void k_rank(const int* __restrict__ wid, const int* __restrict__ chunk_base,
                       const int* __restrict__ offsets, int* __restrict__ dest_row,
                       float* __restrict__ order_out, int M) {
  __shared__ int lw[CHUNK];
  const int t = threadIdx.x;
  const int i = blockIdx.x * CHUNK + t;
  const int myw = (i < NN) ? wid[i] : -1;
  lw[t] = myw;
  __syncthreads();
  if (i >= NN) return;
  int cnt = 0;
  for (int j = 0; j < t; ++j) cnt += (lw[j] == myw) ? 1 : 0;   // stable in-chunk rank
  const int r = chunk_base[blockIdx.x * NW + myw] + cnt;       // global sorted position
  order_out[r] = (float)i;                                     // order[r] = i
  dest_row[i] = myw * M + (r - offsets[myw]);                  // w*M + pos-in-window
}

// ---- Kernel 4: zero only the padding rows ----------------------------------
__global__ void k_zero(const int* __restrict__ counts, float* __restrict__ out, int M) {
  const int w = blockIdx.x / M;
  const int p = blockIdx.x - w * M;
  if (p < counts[w]) return;                       // valid row: scatter will fill it
  const int b = threadIdx.x >> 5;                  // 8 batches x 32 lanes = 256 thr
  const int lane = threadIdx.x & 31;
  const size_t off = (((size_t)b * NW + (size_t)w) * (size_t)M + (size_t)p) * NC
                     + (size_t)lane * 4;
  *reinterpret_cast<float4*>(out + off) = make_float4(0.f, 0.f, 0.f, 0.f);
}

extern "C" void kernel_launch(void* const* d_in, const int* in_sizes, int n_in,
                              void* d_out, int out_size, void* d_ws, size_t ws_size,
                              hipStream_t stream) {
  (void)in_sizes; (void)n_in; (void)ws_size;
  const float* x  = (const float*)d_in[0];
  const int* wid  = (const int*)d_in[1];

  // max_win_size recovered on host from out_size (graph-capture safe)
  const int M = (out_size - NN - NW) / (NB * NW * NC);

  // workspace layout (ints)
  int* ws         = (int*)d_ws;
  int* offsets    = ws;                            // NW
  int* counts     = ws + NW;                       // NW
  int* partial    = ws + 2 * NW;                   // NCHUNKS*NW
  int* chunk_base = partial + NCHUNKS * NW;        // NCHUNKS*NW
  int* dest_row   = chunk_base + NCHUNKS * NW;     // NN

  float* out        = (float*)d_out;
  float* order_out  = out + (size_t)NB * NW * (size_t)M * NC;   // N floats
  float* counts_out = order_out + NN;                           // NW floats

  k_hist<<<dim3(NCHUNKS), dim3(CHUNK), 0, stream>>>(wid, partial);
  k_scan<<<dim3(1), dim3(256), 0, stream>>>(partial, counts, offsets, chunk_base, counts_out);
  k_rank<<<dim3(NCHUNKS), dim3(CHUNK), 0, stream>>>(wid, chunk_base, offsets, dest_row, order_out, M);
  k_zero<<<dim3(NW * M), dim3(256), 0, stream>>>(counts, out, M);
  k_copy<<<dim3((NB * NN) / 8), dim3(256), 0, stream>>>(x, dest_row, out, M);
}